// GATModel_39848706573593
// MI455X (gfx1250) — compile-verified
//
#include <hip/hip_runtime.h>
#include <math.h>

// ---------------------------------------------------------------- constants
#define NN      20000          // nodes
#define NE      320000         // raw edges
#define ETOT    (NE + NN)      // edges + self loops (reference appends loops)
#define IN_CH   128
#define HID     64
#define HEADS   8
#define HC1     (HEADS * HID)  // 512
#define OUT_CH  64

typedef float v2f __attribute__((ext_vector_type(2)));
typedef float v8f __attribute__((ext_vector_type(8)));

// ------------------------------------------------- ordered-int float max map
__device__ __forceinline__ int f2ord(float f) {
    int i = __float_as_int(f);
    return i >= 0 ? i : (i ^ 0x7fffffff);
}
__device__ __forceinline__ float ord2f(int i) {
    return __int_as_float(i >= 0 ? i : (i ^ 0x7fffffff));
}

// ---------------------------------------------------------------- fp32 WMMA GEMM
// C[M,N] = A[M,K] @ B[K,N], row-major. One wave -> 16(M) x 64(N) strip,
// 4 accumulators reuse each A fragment. Requires M%16==0, K%4==0, N%64==0.
// Whole waves exit uniformly -> EXEC all ones at every WMMA (ISA requirement).
// 32-bit unsigned element offsets from uniform bases -> GVS-form global loads.
__global__ void wmma_gemm_f32(const float* __restrict__ A,
                              const float* __restrict__ B,
                              float* __restrict__ C,
                              int M, int K, int N) {
    const int wave    = (blockIdx.x * blockDim.x + threadIdx.x) >> 5;
    const int lane    = threadIdx.x & 31;
    const int nGroups = N >> 6;                  // 64-wide column groups
    const int mTiles  = M >> 4;
    if (wave >= mTiles * nGroups) return;        // uniform per wave
    const int mt   = wave / nGroups;
    const int ng   = wave % nGroups;
    const int l    = lane & 15;                  // row (A) / col (B,C) in tile
    const int half = lane >> 4;                  // K-pair select (0:k0..1, 1:k0+2..3)

    unsigned aoff = (unsigned)(mt * 16 + l) * (unsigned)K + 2u * half;
    unsigned boff = (unsigned)(2 * half) * (unsigned)N + (unsigned)(ng * 64 + l);
    const unsigned uN = (unsigned)N;

    v8f acc0 = {}, acc1 = {}, acc2 = {}, acc3 = {};
    for (int k0 = 0; k0 < K; k0 += 4) {
        v2f af = *(const v2f*)(A + aoff);                  // A[m][k0+2h .. +1]
        v2f bf0 = { B[boff],      B[boff + uN]      };     // B[k0+2h(+1)][n+ 0]
        v2f bf1 = { B[boff + 16], B[boff + uN + 16] };     //              n+16
        v2f bf2 = { B[boff + 32], B[boff + uN + 32] };     //              n+32
        v2f bf3 = { B[boff + 48], B[boff + uN + 48] };     //              n+48
        acc0 = __builtin_amdgcn_wmma_f32_16x16x4_f32(false, af, false, bf0, (short)0, acc0, false, false);
        acc1 = __builtin_amdgcn_wmma_f32_16x16x4_f32(false, af, false, bf1, (short)0, acc1, false, false);
        acc2 = __builtin_amdgcn_wmma_f32_16x16x4_f32(false, af, false, bf2, (short)0, acc2, false, false);
        acc3 = __builtin_amdgcn_wmma_f32_16x16x4_f32(false, af, false, bf3, (short)0, acc3, false, false);
        aoff += 4u;
        boff += 4u * uN;
    }
    // C/D layout: VGPR r, lanes 0-15 -> M=r, lanes 16-31 -> M=r+8; lane%16 -> N
    unsigned coff = (unsigned)(mt * 16 + 8 * half) * uN + (unsigned)(ng * 64 + l);
    #pragma unroll
    for (int r = 0; r < 8; ++r) {
        float* cp = C + coff;
        cp[0]  = acc0[r];
        cp[16] = acc1[r];
        cp[32] = acc2[r];
        cp[48] = acc3[r];
        coff += uN;
    }
}

// ------------------------------------------------ per-(node,head) attention scores
__global__ void scores_kernel(const float* __restrict__ h,
                              const float* __restrict__ a_src,
                              const float* __restrict__ a_dst,
                              float* __restrict__ esrc, float* __restrict__ edst,
                              int* __restrict__ mmax, float* __restrict__ denom,
                              int n, int H, int C) {
    int i = blockIdx.x * blockDim.x + threadIdx.x;   // node*H + head
    if (i >= n * H) return;
    int node = i / H, hh = i % H;
    const float* hp = h + (size_t)node * H * C + (size_t)hh * C;
    const float* as = a_src + hh * C;
    const float* ad = a_dst + hh * C;
    float s = 0.f, d = 0.f;
    for (int c = 0; c < C; ++c) { float v = hp[c]; s += v * as[c]; d += v * ad[c]; }
    esrc[i]  = s;
    edst[i]  = d;
    mmax[i]  = f2ord(-INFINITY);
    denom[i] = 0.f;
}

// ---------------------------------------------------------------- zero fill
__global__ void fill_zero(float* __restrict__ p, long long n) {
    long long i = (long long)blockIdx.x * blockDim.x + threadIdx.x;
    long long stride = (long long)gridDim.x * blockDim.x;
    for (; i < n; i += stride) p[i] = 0.f;
}

// ------------------------------------------- edge pass 1: leaky score + atomic max
__global__ void edge_score_max(const int* __restrict__ ei,
                               const float* __restrict__ esrc,
                               const float* __restrict__ edst,
                               float* __restrict__ eleaky,
                               int* __restrict__ mmax, int H) {
    long long t = (long long)blockIdx.x * blockDim.x + threadIdx.x;
    if (t >= (long long)ETOT * H) return;
    int e = (int)(t / H), hh = (int)(t % H);
    int src, dst;
    if (e < NE) { src = ei[e]; dst = ei[NE + e]; }
    else        { src = dst = e - NE; }                 // self loop
    float v = esrc[src * H + hh] + edst[dst * H + hh];
    v = v > 0.f ? v : 0.2f * v;                         // leaky_relu(0.2)
    eleaky[t] = v;
    atomicMax(&mmax[dst * H + hh], f2ord(v));
}

// ------------------------------------------- edge pass 2: exp + atomic denom sum
__global__ void edge_exp_sum(const int* __restrict__ ei,
                             float* __restrict__ eleaky,   // in: leaky, out: exp
                             const int* __restrict__ mmax,
                             float* __restrict__ denom, int H) {
    long long t = (long long)blockIdx.x * blockDim.x + threadIdx.x;
    if (t >= (long long)ETOT * H) return;
    int e = (int)(t / H), hh = (int)(t % H);
    int dst = (e < NE) ? ei[NE + e] : (e - NE);
    float ex = expf(eleaky[t] - ord2f(mmax[dst * H + hh]));
    eleaky[t] = ex;
    atomicAdd(&denom[dst * H + hh], ex);
}

// ------------------------------------------- edge pass 3: normalize (few divides)
__global__ void edge_alpha(const int* __restrict__ ei,
                           float* __restrict__ exal,       // in: exp, out: alpha
                           const float* __restrict__ denom, int H) {
    long long t = (long long)blockIdx.x * blockDim.x + threadIdx.x;
    if (t >= (long long)ETOT * H) return;
    int e = (int)(t / H), hh = (int)(t % H);
    int dst = (e < NE) ? ei[NE + e] : (e - NE);
    exal[t] = exal[t] / (denom[dst * H + hh] + 1e-16f);
}

// ------------------------------------------- edge pass 4: weighted scatter-add
// thread per (edge, channel): consecutive lanes -> consecutive channels ->
// coalesced gather of h[src] row + coalesced global_atomic_add_f32 to agg[dst].
// alpha precomputed -> this kernel is pure load/fma/atomic (HBM-bound).
__global__ void edge_aggregate(const int* __restrict__ ei,
                               const float* __restrict__ alpha,
                               const float* __restrict__ hval,
                               float* __restrict__ agg, int H, int C) {
    const int HC = H * C;
    long long t = (long long)blockIdx.x * blockDim.x + threadIdx.x;
    if (t >= (long long)ETOT * HC) return;
    int e = (int)(t / HC), c = (int)(t % HC);
    int hh = c / C;
    int src, dst;
    if (e < NE) { src = ei[e]; dst = ei[NE + e]; }
    else        { src = dst = e - NE; }
    float a = alpha[(size_t)e * H + hh];
    atomicAdd(&agg[(size_t)dst * HC + c], a * hval[(size_t)src * HC + c]);
}

// ---------------------------------------------------------------- bias + relu
__global__ void bias_relu(float* __restrict__ a, const float* __restrict__ b,
                          long long n, int F) {
    long long i = (long long)blockIdx.x * blockDim.x + threadIdx.x;
    if (i >= n) return;
    float v = a[i] + b[(int)(i % F)];
    a[i] = v > 0.f ? v : 0.f;
}

// ------------------------------------------- final: + b2, log_softmax (wave32)
__global__ void logsoftmax64(float* __restrict__ out, const float* __restrict__ b2) {
    int node = blockIdx.x;
    int lane = threadIdx.x;                       // 32 threads (one wave)
    float* row = out + (size_t)node * OUT_CH;
    float v0 = row[lane]      + b2[lane];
    float v1 = row[lane + 32] + b2[lane + 32];
    float m = fmaxf(v0, v1);
    #pragma unroll
    for (int o = 16; o > 0; o >>= 1) m = fmaxf(m, __shfl_xor(m, o, 32));
    float s = expf(v0 - m) + expf(v1 - m);
    #pragma unroll
    for (int o = 16; o > 0; o >>= 1) s += __shfl_xor(s, o, 32);
    float lse = logf(s) + m;
    row[lane]      = v0 - lse;
    row[lane + 32] = v1 - lse;
}

// ================================================================ launcher
extern "C" void kernel_launch(void* const* d_in, const int* in_sizes, int n_in,
                              void* d_out, int out_size, void* d_ws, size_t ws_size,
                              hipStream_t stream) {
    const float* x   = (const float*)d_in[0];
    const int*   ei  = (const int*)  d_in[1];
    const float* W1  = (const float*)d_in[2];
    const float* a1s = (const float*)d_in[3];
    const float* a1d = (const float*)d_in[4];
    const float* b1  = (const float*)d_in[5];
    const float* W2  = (const float*)d_in[6];
    const float* a2s = (const float*)d_in[7];
    const float* a2d = (const float*)d_in[8];
    const float* b2  = (const float*)d_in[9];
    float* out = (float*)d_out;

    // -------- workspace carve-up (floats)
    float* ws = (float*)d_ws;
    float* h1pre = ws;                                  // 20000*512
    float* agg1  = h1pre + (size_t)NN * HC1;            // 20000*512 -> h1 after relu
    float* esrc1 = agg1  + (size_t)NN * HC1;            // 20000*8
    float* edst1 = esrc1 + (size_t)NN * HEADS;
    int*   m1    = (int*)(edst1 + (size_t)NN * HEADS);
    float* den1  = (float*)m1 + (size_t)NN * HEADS;
    float* ex1   = den1 + (size_t)NN * HEADS;           // ETOT*8
    float* h2pre = ex1  + (size_t)ETOT * HEADS;         // 20000*64
    float* esrc2 = h2pre + (size_t)NN * OUT_CH;         // 20000
    float* edst2 = esrc2 + NN;
    int*   m2    = (int*)(edst2 + NN);
    float* den2  = (float*)m2 + NN;
    float* ex2   = den2 + NN;                           // ETOT

    const int TB = 256;
    auto blocks = [](long long n, int tb) { return (unsigned)((n + tb - 1) / tb); };

    // ================= layer 1 =================
    // GEMM1: h1pre = x @ W1   (M=20000, K=128, N=512)
    {
        long long waves = (long long)(NN / 16) * (HC1 / 64);    // 10000
        wmma_gemm_f32<<<blocks(waves * 32, TB), TB, 0, stream>>>(x, W1, h1pre, NN, IN_CH, HC1);
    }
    // per-(node,head) scores + init max/denom
    scores_kernel<<<blocks((long long)NN * HEADS, TB), TB, 0, stream>>>(
        h1pre, a1s, a1d, esrc1, edst1, m1, den1, NN, HEADS, HID);
    // zero aggregation target
    fill_zero<<<4096, TB, 0, stream>>>(agg1, (long long)NN * HC1);
    // edge passes
    edge_score_max<<<blocks((long long)ETOT * HEADS, TB), TB, 0, stream>>>(ei, esrc1, edst1, ex1, m1, HEADS);
    edge_exp_sum  <<<blocks((long long)ETOT * HEADS, TB), TB, 0, stream>>>(ei, ex1, m1, den1, HEADS);
    edge_alpha    <<<blocks((long long)ETOT * HEADS, TB), TB, 0, stream>>>(ei, ex1, den1, HEADS);
    edge_aggregate<<<blocks((long long)ETOT * HC1,   TB), TB, 0, stream>>>(ei, ex1, h1pre, agg1, HEADS, HID);
    // h1 = relu(agg1 + b1)  (in place)
    bias_relu<<<blocks((long long)NN * HC1, TB), TB, 0, stream>>>(agg1, b1, (long long)NN * HC1, HC1);

    // ================= layer 2 =================
    // GEMM2: h2pre = h1 @ W2   (M=20000, K=512, N=64)
    {
        long long waves = (long long)(NN / 16) * (OUT_CH / 64); // 1250
        wmma_gemm_f32<<<blocks(waves * 32, TB), TB, 0, stream>>>(agg1, W2, h2pre, NN, HC1, OUT_CH);
    }
    scores_kernel<<<blocks((long long)NN, TB), TB, 0, stream>>>(
        h2pre, a2s, a2d, esrc2, edst2, m2, den2, NN, 1, OUT_CH);
    fill_zero<<<2048, TB, 0, stream>>>(out, (long long)NN * OUT_CH);
    edge_score_max<<<blocks((long long)ETOT, TB), TB, 0, stream>>>(ei, esrc2, edst2, ex2, m2, 1);
    edge_exp_sum  <<<blocks((long long)ETOT, TB), TB, 0, stream>>>(ei, ex2, m2, den2, 1);
    edge_alpha    <<<blocks((long long)ETOT, TB), TB, 0, stream>>>(ei, ex2, den2, 1);
    edge_aggregate<<<blocks((long long)ETOT * OUT_CH, TB), TB, 0, stream>>>(ei, ex2, h2pre, out, 1, OUT_CH);
    // mean over 1 head == identity; add b2 + log_softmax in place
    logsoftmax64<<<NN, 32, 0, stream>>>(out, b2);
}